// GPCALayer_30107720745358
// MI455X (gfx1250) — compile-verified
//
#include <hip/hip_runtime.h>

#define NF 64          // feature width (NIN == NOUT == 64)
#define NC 40          // num classes
#define NPOW 50
#define A_COEF 0.5f    // ALPHA/(1+ALPHA), ALPHA=1
#define B_COEF 0.5f    // 1/(1+ALPHA)
#define BETA 0.1f

typedef __attribute__((ext_vector_type(2))) float v2f;
typedef __attribute__((ext_vector_type(8))) float v8f;

// ---------------- utility ----------------
__global__ void zero_u32_kernel(unsigned* __restrict__ p, long n) {
  long g = (long)blockIdx.x * blockDim.x + threadIdx.x;
  if (g < n) p[g] = 0u;
}

// ---------------- setup: degrees & CSR ----------------
__global__ void deg_count_kernel(const int* __restrict__ row, int* __restrict__ degcnt, int E) {
  int g = blockIdx.x * blockDim.x + threadIdx.x;
  if (g < E) atomicAdd(&degcnt[row[g]], 1);
}

__global__ void inv_deg_kernel(const int* __restrict__ degcnt, float* __restrict__ inv_deg, int n) {
  int g = blockIdx.x * blockDim.x + threadIdx.x;
  if (g < n) inv_deg[g] = 1.0f / ((float)degcnt[g] + 1.0f);
}

// single-block exclusive scan of degcnt -> row_ptr[0..n] (1024 threads)
__global__ void scan_kernel(const int* __restrict__ cnt, int* __restrict__ ptr, int n) {
  __shared__ int part[1024];
  int tid = threadIdx.x;
  int chunk = (n + 1023) >> 10;
  int beg = tid * chunk;
  int end = beg + chunk; if (end > n) end = n;
  int s = 0;
  for (int i = beg; i < end; ++i) s += cnt[i];
  part[tid] = s;
  __syncthreads();
  for (int off = 1; off < 1024; off <<= 1) {
    int t = 0;
    if (tid >= off) t = part[tid - off];
    __syncthreads();
    if (tid >= off) part[tid] += t;
    __syncthreads();
  }
  int run = (tid == 0) ? 0 : part[tid - 1];   // exclusive prefix of this chunk
  for (int i = beg; i < end; ++i) { ptr[i] = run; run += cnt[i]; }
  if (tid == 1023) ptr[n] = part[1023];
}

// scatter col indices into CSR order (cursor must be zeroed)
__global__ void scatter_kernel(const int* __restrict__ row, const int* __restrict__ col,
                               const int* __restrict__ ptr, int* __restrict__ cursor,
                               int* __restrict__ col_sorted, int E) {
  int g = blockIdx.x * blockDim.x + threadIdx.x;
  if (g >= E) return;
  int r = row[g];
  int pos = atomicAdd(&cursor[r], 1);
  col_sorted[ptr[r] + pos] = col[g];
}

// ---------------- setup: label norms ----------------
__global__ void class_count_kernel(const int* __restrict__ y, const int* __restrict__ tm,
                                   int* __restrict__ ccnt, int n) {
  int g = blockIdx.x * blockDim.x + threadIdx.x;
  if (g < n && tm[g] > 0) atomicAdd(&ccnt[y[g]], 1);
}

__global__ void yyt_kernel(const int* __restrict__ y, const int* __restrict__ tm,
                           const int* __restrict__ ccnt, float* __restrict__ yyt, int n) {
  int g = blockIdx.x * blockDim.x + threadIdx.x;
  if (g < n) yyt[g] = ((tm[g] > 0) ? (float)ccnt[y[g]] : 0.0f) + 1e-8f;
}

// ---------------- setup: center x ----------------
__global__ void colsum_kernel(const float* __restrict__ x, float* __restrict__ colsum, int n) {
  int f = threadIdx.x & 63;
  int sub = blockIdx.x * (blockDim.x >> 6) + (threadIdx.x >> 6);
  int stride = gridDim.x * (blockDim.x >> 6);
  float s = 0.0f;
  for (int i = sub; i < n; i += stride) s += x[(long)i * NF + f];
  atomicAdd(&colsum[f], s);
}

__global__ void center_kernel(const float* __restrict__ x, const float* __restrict__ colsum,
                              float* __restrict__ xc, long total, float invn) {
  long g = (long)blockIdx.x * blockDim.x + threadIdx.x;
  if (g < total) xc[g] = x[g] - colsum[g & 63] * invn;
}

// ---------------- per-iteration kernels ----------------
// class_sum[c][f] = sum over train nodes with y==c of v[i][f]  (LDS-tiled)
__global__ void class_sum_kernel(const float* __restrict__ v, const int* __restrict__ y,
                                 const int* __restrict__ tm, float* __restrict__ classsum, int n) {
  __shared__ float sh[NC * NF];
  int tid = threadIdx.x;                 // 256 threads
  for (int i = tid; i < NC * NF; i += 256) sh[i] = 0.0f;
  __syncthreads();
  int f  = tid & 63;
  int rl = tid >> 6;                     // 0..3
  int base = blockIdx.x * 512;
  int end  = base + 512; if (end > n) end = n;
  for (int i = base + rl; i < end; i += 4) {
    if (tm[i] > 0) atomicAdd(&sh[y[i] * NF + f], v[(long)i * NF + f]);
  }
  __syncthreads();
  for (int i = tid; i < NC * NF; i += 256) atomicAdd(&classsum[i], sh[i]);
}

// Fused propagate: one wave per node.
// vnext[r] = a*((1-B)*inv_deg[r]*(sum_{c in CSR[r]} v[c] + v[r]) + B*p2[r]) + b*xc[r]
// Edge columns are loaded coalesced 32-at-a-time and broadcast via shfl.
__global__ void propagate_kernel(const float* __restrict__ v, const int* __restrict__ ptr,
                                 const int* __restrict__ cols, const float* __restrict__ inv_deg,
                                 const float* __restrict__ classsum, const int* __restrict__ y,
                                 const int* __restrict__ tm, const float* __restrict__ yyt,
                                 const float* __restrict__ xc, float* __restrict__ vnext, int n) {
  int wave = (blockIdx.x * blockDim.x + threadIdx.x) >> 5;
  int lane = threadIdx.x & 31;
  if (wave >= n) return;                 // wave-uniform guard
  const int r = wave;
  const int beg = ptr[r];
  const int end = ptr[r + 1];

  float ax = 0.0f, ay = 0.0f;
  for (int base = beg; base < end; base += 32) {
    int nload = end - base; if (nload > 32) nload = 32;
    int cl = (lane < nload) ? cols[base + lane] : 0;   // coalesced chunk load
    for (int k = 0; k < nload; ++k) {
      int c = __shfl(cl, k, 32);
      const float2 t = *(const float2*)(v + (long)c * NF + lane * 2);
      ax += t.x;
      ay += t.y;
    }
  }

  const long gbase = (long)r * NF + lane * 2;
  const float2 sv = *(const float2*)(v + gbase);       // self-loop term
  const float idr = inv_deg[r];
  const float p1x = (ax + sv.x) * idr;
  const float p1y = (ay + sv.y) * idr;

  float p2x = 0.0f, p2y = 0.0f;
  if (tm[r] > 0) {
    const float iy = 1.0f / yyt[r];
    const float2 cs = *(const float2*)(classsum + y[r] * NF + lane * 2);
    p2x = cs.x * iy;
    p2y = cs.y * iy;
  }

  const float2 xcv = *(const float2*)(xc + gbase);
  float2 outv;
  outv.x = A_COEF * ((1.0f - BETA) * p1x + BETA * p2x) + B_COEF * xcv.x;
  outv.y = A_COEF * ((1.0f - BETA) * p1y + BETA * p2y) + B_COEF * xcv.y;
  *(float2*)(vnext + gbase) = outv;
}

// ---------------- final projection: out = V @ W + bias, via V_WMMA_F32_16X16X4_F32 ----------------
// One wave computes one 16x16 output tile; 16 WMMA steps over K=64.
// A (16x4 f32): lane m = lane&15 holds K = 2*(lane>>4)+{0,1}
// B (4x16 f32): lane n = lane&15; VGPR j holds K = 2*(lane>>4)+j
// C/D (16x16 f32): VGPR r -> M = r + 8*(lane>>4), N = lane&15
__global__ void gemm_wmma_kernel(const float* __restrict__ V, const float* __restrict__ W,
                                 const float* __restrict__ bias, float* __restrict__ out, int n) {
  int wave = (blockIdx.x * blockDim.x + threadIdx.x) >> 5;
  int lane = threadIdx.x & 31;
  int tiles_m = (n + 15) >> 4;
  int tm = wave >> 2;            // NOUT/16 == 4 tiles in N
  int tn = wave & 3;
  if (tm >= tiles_m) return;     // wave-uniform guard: EXEC stays all-ones for WMMA
  int m0 = tm << 4;
  int n0 = tn << 4;
  int m  = lane & 15;
  int kh = lane >> 4;            // 0 or 1

  int arow = m0 + m; if (arow > n - 1) arow = n - 1;   // clamp (n is a multiple of 16 anyway)

  v8f acc = {};
#pragma unroll
  for (int kk = 0; kk < NF; kk += 4) {
    int k0 = kk + 2 * kh;
    v2f a, b;
    a.x = V[(long)arow * NF + k0 + 0];
    a.y = V[(long)arow * NF + k0 + 1];
    b.x = W[(long)(k0 + 0) * NF + n0 + m];
    b.y = W[(long)(k0 + 1) * NF + n0 + m];
    acc = __builtin_amdgcn_wmma_f32_16x16x4_f32(false, a, false, b, (short)0, acc, false, false);
  }
  float bv = bias[n0 + m];
#pragma unroll
  for (int r = 0; r < 8; ++r) {
    int orow = m0 + r + 8 * kh;
    if (orow < n) out[(long)orow * NF + n0 + m] = acc[r] + bv;
  }
}

// ---------------- host ----------------
extern "C" void kernel_launch(void* const* d_in, const int* in_sizes, int n_in,
                              void* d_out, int out_size, void* d_ws, size_t ws_size,
                              hipStream_t stream) {
  const float* x    = (const float*)d_in[0];
  const float* W    = (const float*)d_in[1];
  const float* bias = (const float*)d_in[2];
  const int*   row  = (const int*)d_in[3];
  const int*   col  = (const int*)d_in[4];
  const int*   y    = (const int*)d_in[5];
  const int*   tmsk = (const int*)d_in[6];

  const int n = in_sizes[0] / NF;
  const int E = in_sizes[3];
  const long total = (long)n * NF;

  // workspace carve-out (256B aligned slices)
  char* ws = (char*)d_ws;
  size_t off = 0;
  auto carve = [&](size_t bytes) -> char* {
    char* p = ws + off;
    off += (bytes + 255) & ~(size_t)255;
    return p;
  };
  float* inv_deg    = (float*)carve((size_t)n * 4);
  float* yyt        = (float*)carve((size_t)n * 4);
  float* xc         = (float*)carve((size_t)total * 4);
  float* v0         = (float*)carve((size_t)total * 4);
  float* v1         = (float*)carve((size_t)total * 4);
  float* classsum   = (float*)carve((size_t)NC * NF * 4);
  float* colsum     = (float*)carve(64 * 4);
  int*   degcnt     = (int*)carve((size_t)n * 4);
  int*   ccnt       = (int*)carve((size_t)NC * 4);
  int*   row_ptr    = (int*)carve((size_t)(n + 1) * 4);
  int*   cursor     = (int*)carve((size_t)n * 4);
  int*   col_sorted = (int*)carve((size_t)E * 4);

  const int T = 256;
  dim3 blk(T);

  // ---- setup: degrees + CSR build ----
  zero_u32_kernel<<<(n + T - 1) / T, blk, 0, stream>>>((unsigned*)degcnt, n);
  zero_u32_kernel<<<(n + T - 1) / T, blk, 0, stream>>>((unsigned*)cursor, n);
  deg_count_kernel<<<(E + T - 1) / T, blk, 0, stream>>>(row, degcnt, E);
  inv_deg_kernel<<<(n + T - 1) / T, blk, 0, stream>>>(degcnt, inv_deg, n);
  scan_kernel<<<1, 1024, 0, stream>>>(degcnt, row_ptr, n);
  scatter_kernel<<<(E + T - 1) / T, blk, 0, stream>>>(row, col, row_ptr, cursor, col_sorted, E);

  // ---- setup: label norms ----
  zero_u32_kernel<<<1, blk, 0, stream>>>((unsigned*)ccnt, NC);
  class_count_kernel<<<(n + T - 1) / T, blk, 0, stream>>>(y, tmsk, ccnt, n);
  yyt_kernel<<<(n + T - 1) / T, blk, 0, stream>>>(y, tmsk, ccnt, yyt, n);

  // ---- setup: center x ----
  zero_u32_kernel<<<1, blk, 0, stream>>>((unsigned*)colsum, 64);
  colsum_kernel<<<256, blk, 0, stream>>>(x, colsum, n);
  center_kernel<<<(int)((total + T - 1) / T), blk, 0, stream>>>(x, colsum, xc, total, 1.0f / (float)n);

  // ---- power iterations (ping-pong, fused SpMM+combine) ----
  const float* vin = xc;   // scan starts from xc
  float* vout = v0;
  const int prop_blocks = (int)(((long)n * 32 + T - 1) / T);
  for (int it = 0; it < NPOW; ++it) {
    zero_u32_kernel<<<(NC * NF + T - 1) / T, blk, 0, stream>>>((unsigned*)classsum, NC * NF);
    class_sum_kernel<<<(n + 511) / 512, blk, 0, stream>>>(vin, y, tmsk, classsum, n);
    propagate_kernel<<<prop_blocks, blk, 0, stream>>>(vin, row_ptr, col_sorted, inv_deg,
                                                      classsum, y, tmsk, yyt, xc, vout, n);
    vin = vout;
    vout = (vout == v0) ? v1 : v0;
  }

  // ---- final projection via WMMA f32 ----
  int tiles_m = (n + 15) / 16;
  long waves = (long)tiles_m * 4;
  long threads = waves * 32;
  gemm_wmma_kernel<<<(int)((threads + T - 1) / T), blk, 0, stream>>>(vin, W, bias, (float*)d_out, n);
}